// LabelSwitch_6313601925366
// MI455X (gfx1250) — compile-verified
//
#include <hip/hip_runtime.h>
#include <hip/hip_bf16.h>
#include <limits.h>

#ifndef __has_builtin
#define __has_builtin(x) 0
#endif

constexpr int L = 1000;            // NUM_LABELS
constexpr int NROWS = 64 * 512;    // 32768
constexpr int ROWS_PER_BLOCK = 8;  // gather staging depth

// ---------------------------------------------------------------------------
// CDNA5 async global->LDS helpers (ASYNCcnt-tracked data mover path)
// ---------------------------------------------------------------------------
typedef int v4i_t __attribute__((vector_size(16)));
typedef __attribute__((address_space(1))) v4i_t* gas_v4i_ptr;
typedef __attribute__((address_space(3))) v4i_t* lds_v4i_ptr;

__device__ __forceinline__ void async_copy_b128(const float* g, float* l) {
#if __has_builtin(__builtin_amdgcn_global_load_async_to_lds_b128)
  __builtin_amdgcn_global_load_async_to_lds_b128(
      (gas_v4i_ptr)g, (lds_v4i_ptr)l, 0, 0);
#else
  // LDS generic address keeps the byte offset in the low 32 bits.
  unsigned lds_off = (unsigned)(unsigned long long)(void*)l;
  unsigned long long ga = (unsigned long long)(const void*)g;
  asm volatile("global_load_async_to_lds_b128 %0, %1, off"
               :: "v"(lds_off), "v"(ga) : "memory");
#endif
}

__device__ __forceinline__ void wait_asynccnt0() {
#if __has_builtin(__builtin_amdgcn_s_wait_asynccnt)
  __builtin_amdgcn_s_wait_asynccnt(0);
#else
  asm volatile("s_wait_asynccnt 0" ::: "memory");
#endif
}

// ---------------------------------------------------------------------------
// K0: init workspace state (firstRow = INT_MAX, count = 0) every call
// ---------------------------------------------------------------------------
__global__ void k_init(int* __restrict__ firstRow, int* __restrict__ count) {
  int t = threadIdx.x;
  if (t < 1024) firstRow[t] = INT_MAX;
  if (t == 0) *count = 0;
}

// ---------------------------------------------------------------------------
// K1: per-row argmax (first-occurrence tie break, matching jnp.argmax) and
//     first-occurrence row per label via atomicMin. One block per row.
// ---------------------------------------------------------------------------
__global__ void k_rowargmax(const float* __restrict__ x,
                            const int* __restrict__ labels,
                            int* __restrict__ rowArg,
                            int* __restrict__ firstRow) {
  __shared__ float sv[256];
  __shared__ int   si[256];
  const int row = blockIdx.x;
  const int t   = threadIdx.x;
  const float* p = x + (size_t)row * L;

  float bv = -__builtin_huge_valf();
  int   bi = 0x7fffffff;
  if (t < 250) {  // 250 * float4 == 1000 elements, rows are 16B aligned (4000B)
    float4 v = ((const float4*)p)[t];
    int j = 4 * t;
    bv = v.x; bi = j;
    if (v.y > bv) { bv = v.y; bi = j + 1; }
    if (v.z > bv) { bv = v.z; bi = j + 2; }
    if (v.w > bv) { bv = v.w; bi = j + 3; }
  }
  sv[t] = bv; si[t] = bi;
  __syncthreads();
  for (int s = 128; s > 0; s >>= 1) {
    if (t < s) {
      float ov = sv[t + s]; int oi = si[t + s];
      if (ov > sv[t] || (ov == sv[t] && oi < si[t])) { sv[t] = ov; si[t] = oi; }
    }
    __syncthreads();
  }
  if (t == 0) {
    rowArg[row] = si[0];
    atomicMin(&firstRow[labels[row]], row);  // only first occurrence matters
  }
}

// ---------------------------------------------------------------------------
// K2: build the compact effective-step sequence, ordered by first-occurrence
//     row. Ranks are unique since firstRow values are distinct.
// ---------------------------------------------------------------------------
__global__ void k_rank(const int* __restrict__ firstRow,
                       const int* __restrict__ rowArg,
                       int* __restrict__ seq_m,
                       int* __restrict__ seq_lab,
                       int* __restrict__ count) {
  __shared__ int fr[1024];
  int t = threadIdx.x;  // blockDim == 1024
  fr[t] = (t < L) ? firstRow[t] : INT_MAX;
  __syncthreads();
  if (t < L) {
    int f = fr[t];
    if (f != INT_MAX) {
      int rank = 0;
      for (int l = 0; l < L; ++l) rank += (fr[l] < f) ? 1 : 0;
      seq_m[rank]   = rowArg[f];  // true argmax of that row
      seq_lab[rank] = t;          // the label
      atomicAdd(count, 1);
    }
  }
}

// ---------------------------------------------------------------------------
// K3: the (collapsed) serial scan: <=1000 O(1) steps in LDS.
//     r = inv[m]; if r != lab, apply transposition (r<->lab) to sel values.
// ---------------------------------------------------------------------------
__global__ void k_scan(const int* __restrict__ seq_m,
                       const int* __restrict__ seq_lab,
                       const int* __restrict__ count,
                       int* __restrict__ perm) {
  __shared__ int inv[L];
  __shared__ int sel[L];
  int t = threadIdx.x;  // blockDim == 1024
  if (t < L) { inv[t] = t; sel[t] = t; }
  __syncthreads();
  if (t == 0) {
    int c = *count;
    for (int k = 0; k < c; ++k) {
      int m = seq_m[k], lab = seq_lab[k];
      int r = inv[m];                 // argmax position in permuted space
      if (r != lab) {
        int a = inv[r], b = inv[lab];
        sel[a] = lab; sel[b] = r;
        inv[lab] = a; inv[r] = b;
      }
    }
  }
  __syncthreads();
  if (t < L) perm[t] = sel[t];
}

// ---------------------------------------------------------------------------
// K4: apply the final column permutation to every row.
//     Async-stage ROWS_PER_BLOCK rows into LDS (global_load_async_to_lds_b128),
//     gather from LDS, write coalesced float4 stores. Reads should mostly hit
//     the 192MB L2 warmed by K1 (tensor is 131MB).
// ---------------------------------------------------------------------------
__global__ void k_gather(const float* __restrict__ x,
                         const int* __restrict__ perm,
                         float* __restrict__ out) {
  __shared__ int   sperm[1024];
  __shared__ float srow[ROWS_PER_BLOCK][1024];  // 1024-float stride keeps 16B align
  const int t = threadIdx.x;  // blockDim == 256

  for (int j = t; j < L; j += 256) sperm[j] = perm[j];

  const size_t rbase = (size_t)blockIdx.x * ROWS_PER_BLOCK;
  if (t < 250) {
    for (int rr = 0; rr < ROWS_PER_BLOCK; ++rr) {
      const float* g = x + (rbase + rr) * L + 4 * t;  // 16B aligned (4000B rows)
      async_copy_b128(g, &srow[rr][4 * t]);
    }
  }
  wait_asynccnt0();   // issuing wave's LDS writes complete
  __syncthreads();    // publish to all waves in the block

  if (t < 250) {
    const int j = 4 * t;
    const int p0 = sperm[j], p1 = sperm[j + 1], p2 = sperm[j + 2], p3 = sperm[j + 3];
    for (int rr = 0; rr < ROWS_PER_BLOCK; ++rr) {
      float4 o;
      o.x = srow[rr][p0];
      o.y = srow[rr][p1];
      o.z = srow[rr][p2];
      o.w = srow[rr][p3];
      ((float4*)(out + (rbase + rr) * L))[t] = o;
    }
  }
}

// ---------------------------------------------------------------------------
// Launch
// ---------------------------------------------------------------------------
extern "C" void kernel_launch(void* const* d_in, const int* in_sizes, int n_in,
                              void* d_out, int out_size, void* d_ws, size_t ws_size,
                              hipStream_t stream) {
  const float* x      = (const float*)d_in[0];  // outputs [64,512,1000] f32
  const int*   labels = (const int*)d_in[1];    // labels  [64,512] i32
  float*       out    = (float*)d_out;

  char* ws = (char*)d_ws;
  int* firstRow = (int*)(ws);                          // 4KB
  int* rowArg   = (int*)(ws + 4096);                   // 128KB
  int* seq_m    = (int*)(ws + 4096 + 131072);          // 4KB
  int* seq_lab  = (int*)(ws + 4096 + 131072 + 4096);   // 4KB
  int* count    = (int*)(ws + 4096 + 131072 + 8192);   // 4B (padded)
  int* perm     = (int*)(ws + 4096 + 131072 + 12288);  // 4KB

  k_init     <<<1,          1024, 0, stream>>>(firstRow, count);
  k_rowargmax<<<NROWS,       256, 0, stream>>>(x, labels, rowArg, firstRow);
  k_rank     <<<1,          1024, 0, stream>>>(firstRow, rowArg, seq_m, seq_lab, count);
  k_scan     <<<1,          1024, 0, stream>>>(seq_m, seq_lab, count, perm);
  k_gather   <<<NROWS / ROWS_PER_BLOCK, 256, 0, stream>>>(x, perm, out);
}